// TVB_30983894073957
// MI455X (gfx1250) — compile-verified
//
#include <hip/hip_runtime.h>
#include <math.h>

// ---------------------------------------------------------------------------
// TVB MPR simulation, persistent single-WGP kernel for gfx1250 (MI455X).
//  - 256 threads (8 wave32) in ONE block; 1000-step serial Heun scan.
//  - r-history delay ring in LDS (256 x 128, pitch 129 => conflict-free),
//    ~129 KB: only possible with CDNA5's 320KB/WGP LDS.
//  - Delayed coupling row-sums on the matrix pipe via V_WMMA_F32_16X16X4_F32
//    with an all-ones B matrix: D = (W .* R_delayed) x 1.
//  - KEY: cx1(T+1) == cx2(T) (identical gather times, immutable history), so
//    only ONE coupling window is computed per step and ping-ponged.
//  - Noise prefetched one step ahead; BOLD fused in registers.
// ---------------------------------------------------------------------------

typedef float v2f __attribute__((ext_vector_type(2)));
typedef float v8f __attribute__((ext_vector_type(8)));

#if defined(__has_builtin)
#  if __has_builtin(__builtin_amdgcn_wmma_f32_16x16x4_f32)
#    define HAS_WMMA4 1
#  endif
#endif
#ifndef HAS_WMMA4
#  define HAS_WMMA4 0
#endif

namespace {
constexpr int   kN         = 128;
constexpr int   kRingT     = 256;   // power of two >= NH+1 (NH=128)
constexpr int   kRingPitch = 129;   // 128 + 1 pad -> bank = (t + j) % 64
constexpr int   kSteps     = 1000;  // NCHUNK * STEPS
constexpr float kDT        = 0.1f;
constexpr float kSig       = 0.0031622776601683794f;  // SIGMA * sqrt(DT)
constexpr float kPi        = 3.14159265358979323846f;
constexpr float kInvPi     = 0.3183098861837907f;     // DELTA/(pi*TAU)
constexpr float kJC        = 15.0f;                   // J_C * TAU
// BOLD constants
constexpr float kBDT   = 0.01f;
constexpr float kRTS   = 1.0f / 0.65f;
constexpr float kRTF   = 1.0f / 0.41f;
constexpr float kRTO   = 1.0f / 0.98f;
constexpr float kRAl   = 3.125f;          // 1/0.32
constexpr float kInvE0 = 2.5f;            // 1/E0
constexpr float kOneMinusE0 = 0.6f;
constexpr float kV0 = 4.0f;
constexpr float kK1 = 4.3f * 40.3f * 0.4f * 0.04f;
constexpr float kK2 = 0.5f * 25.0f * 0.4f * 0.04f;
constexpr float kK3 = 0.5f;
}  // namespace

__global__ __launch_bounds__(256) void tvb_mpr_persistent(
    const float* __restrict__ region_pars,   // (128,1)
    const float* __restrict__ Wt,            // (128,128,1)
    const float* __restrict__ g,             // (1,)
    const float* __restrict__ stimulus,      // (100,1)
    const float* __restrict__ noise,         // (1000,128,2) flat
    const float* __restrict__ initial_cond,  // (128,2)
    const int*   __restrict__ lags,          // (128,128), values in [1,128]
    float* __restrict__ out)                 // rv (1000,128,2) ++ bold (128)
{
  extern __shared__ float smem[];
  float* ring  = smem;                            // kRingT * kRingPitch
  float* cxA   = ring + kRingT * kRingPitch;      // 128 (window buffer A)
  float* cxB   = cxA + kN;                        // 128 (window buffer B)
  float* stiml = cxB + kN;                        // 100 (+pad)

  const int tid   = threadIdx.x;
  const int lane  = tid & 31;
  const int wav   = tid >> 5;           // 0..7, each wave owns 16 rows
  const int mrow  = 16 * wav + (lane & 15);
  const int cbase = (lane >> 4) << 1;   // K offset within chunk: 0 or 2

  // Per-lane W and pre-transformed lag offsets for its 64 (row,col) pairs.
  // slot(T) for the "window at T+1" gather is (T + q) & 255, q = (1-lag)&255.
  float wreg[64];
  int   qreg[64];
#pragma unroll
  for (int kt = 0; kt < 32; ++kt) {
    const int j0 = 4 * kt + cbase;
    wreg[2 * kt + 0] = Wt[mrow * kN + j0];
    wreg[2 * kt + 1] = Wt[mrow * kN + j0 + 1];
    qreg[2 * kt + 0] = (1 - lags[mrow * kN + j0]) & 255;
    qreg[2 * kt + 1] = (1 - lags[mrow * kN + j0 + 1]) & 255;
  }

  // Stimulus table (one value per 10-step chunk) into LDS.
  for (int t = tid; t < 100; t += 256) stiml[t] = stimulus[t];
  // History ring: every slot starts as initial r (history at t<=0 == initial).
  for (int idx = tid; idx < kRingT * kN; idx += 256) {
    const int s = idx >> 7, j = idx & 127;
    ring[s * kRingPitch + j] = initial_cond[2 * j];
  }

  float cur_r = 0.f, cur_V = 0.f, eta = 0.f;
  float bs = 1.f, bfl = 1.f, bvl = 1.f, bq = 1.f;  // BOLD s,f,v,q
  const float g0 = g[0];
  float2 nz_cur; nz_cur.x = 0.f; nz_cur.y = 0.f;
  if (tid < kN) {
    cur_r  = initial_cond[2 * tid];
    cur_V  = initial_cond[2 * tid + 1];
    eta    = region_pars[tid];
    nz_cur = ((const float2*)noise)[tid];   // prefetch noise for T=0
  }
  __syncthreads();

  // Coupling window: cx_out[i] = sum_j W[i,j] * r(base+1 - lag[i,j], j),
  // i.e. gathers ring slot (base + q) & 255. Two independent WMMA chains.
  auto coupling_window = [&](int base, float* cx_out) {
    v8f accE = {};
    v8f accO = {};
#if HAS_WMMA4
    const v2f bones = {1.0f, 1.0f};
#endif
#pragma unroll
    for (int kt = 0; kt < 32; ++kt) {
      const int j0 = 4 * kt + cbase;
      const int t0 = (base + qreg[2 * kt + 0]) & 255;
      const int t1 = (base + qreg[2 * kt + 1]) & 255;
      v2f a;
      a.x = wreg[2 * kt + 0] * ring[t0 * kRingPitch + j0];
      a.y = wreg[2 * kt + 1] * ring[t1 * kRingPitch + j0 + 1];
#if HAS_WMMA4
      if (kt & 1) {
        accO = __builtin_amdgcn_wmma_f32_16x16x4_f32(
            false, a, false, bones, (short)0, accO, false, false);
      } else {
        accE = __builtin_amdgcn_wmma_f32_16x16x4_f32(
            false, a, false, bones, (short)0, accE, false, false);
      }
#else
      if (kt & 1) accO[kt & 7] += a.x + a.y;
      else        accE[kt & 7] += a.x + a.y;
#endif
    }
#if HAS_WMMA4
    const v8f acc = accE + accO;  // D tiles add elementwise
    // Row sum replicated over N: lane 0 holds rows 16w+0..7, lane 16 holds
    // rows 16w+8..15 (32-bit 16x16 C/D layout).
    if (lane == 0) {
#pragma unroll
      for (int v = 0; v < 8; ++v) cx_out[16 * wav + v] = acc[v];
    } else if (lane == 16) {
#pragma unroll
      for (int v = 0; v < 8; ++v) cx_out[16 * wav + 8 + v] = acc[v];
    }
#else
    float p = accE[0] + accE[1] + accE[2] + accE[3] +
              accE[4] + accE[5] + accE[6] + accE[7] +
              accO[0] + accO[1] + accO[2] + accO[3] +
              accO[4] + accO[5] + accO[6] + accO[7];
    p += __shfl_xor(p, 16, 32);
    if (lane < 16) cx_out[16 * wav + lane] = p;
#endif
  };

  // Prologue: window at times (0 - lag) == cx1 for step 0 (all-init history).
  coupling_window(-1, cxA);

  for (int T = 0; T < kSteps; ++T) {
    float* const cxP = (T & 1) ? cxB : cxA;  // prev window == cx1(T)
    float* const cxN = (T & 1) ? cxA : cxB;  // next window == cx2(T)

    // ---- phase A: publish r_T into ring; prefetch noise for T+1 -----------
    float2 nz_nxt; nz_nxt.x = 0.f; nz_nxt.y = 0.f;
    if (tid < kN) {
      ring[(T & 255) * kRingPitch + tid] = cur_r;
      if (T + 1 < kSteps) nz_nxt = ((const float2*)noise)[(T + 1) * kN + tid];
    }
    __syncthreads();

    // ---- phase B: one coupling window (times T+1-lag) on the matrix pipe --
    coupling_window(T, cxN);
    __syncthreads();

    // ---- phase C: per-region Heun update + rv store + fused BOLD ----------
    if (tid < kN) {
      const float dwr = kSig * nz_cur.x;
      const float dwv = kSig * nz_cur.y;
      const float stv = (tid == 0) ? stiml[T / 10] : 0.f;

      const float rc = cur_r, Vc = cur_V;
      const float I1 = g0 * cxP[tid] + stv;
      const float d1r = kInvPi + 2.f * rc * Vc;
      const float pt1 = kPi * rc;
      const float d1V = Vc * Vc + eta + kJC * rc + I1 - pt1 * pt1;
      const float xir = fmaxf(rc + kDT * d1r + dwr, 0.f);
      const float xiV = Vc + kDT * d1V + dwv;

      const float I2 = g0 * cxN[tid] + stv;
      const float d2r = kInvPi + 2.f * xir * xiV;
      const float pt2 = kPi * xir;
      const float d2V = xiV * xiV + eta + kJC * xir + I2 - pt2 * pt2;

      const float nxr = fmaxf(rc + kDT * 0.5f * (d1r + d2r) + dwr, 0.f);
      const float nxV = Vc + kDT * 0.5f * (d1V + d2V) + dwv;
      cur_r = nxr;
      cur_V = nxV;

      float2 rv2; rv2.x = nxr; rv2.y = nxV;
      ((float2*)out)[T * kN + tid] = rv2;

      // BOLD Heun step with x = nxr
      {
        const float x   = nxr;
        const float vr1 = powf(bvl, kRAl);
        const float d1s = x - kRTS * bs - kRTF * (bfl - 1.f);
        const float d1f = bs;
        const float d1v = kRTO * (bfl - vr1);
        const float d1q = kRTO * (bfl * (1.f - powf(kOneMinusE0, 1.f / bfl)) * kInvE0
                                  - vr1 * bq / bvl);
        const float s2 = bs + kBDT * d1s;
        const float f2 = bfl + kBDT * d1f;
        const float v2 = bvl + kBDT * d1v;
        const float q2 = bq + kBDT * d1q;
        const float vr2 = powf(v2, kRAl);
        const float d2s = x - kRTS * s2 - kRTF * (f2 - 1.f);
        const float d2f = s2;
        const float d2v = kRTO * (f2 - vr2);
        const float d2q = kRTO * (f2 * (1.f - powf(kOneMinusE0, 1.f / f2)) * kInvE0
                                  - vr2 * q2 / v2);
        bs  += 0.5f * kBDT * (d1s + d2s);
        bfl += 0.5f * kBDT * (d1f + d2f);
        bvl += 0.5f * kBDT * (d1v + d2v);
        bq  += 0.5f * kBDT * (d1q + d2q);
      }
      nz_cur = nz_nxt;
    }
    // No barrier needed here: next phase A only writes ring slot (T+1)&255
    // (stale by 256 steps, never gathered) and this thread's own registers;
    // the old-prev cx buffer is rewritten only after the next phase-A sync.
  }

  if (tid < kN) {
    const float bold = kV0 * (kK1 * (1.f - bq) + kK2 * (1.f - bq / bvl) +
                              kK3 * (1.f - bvl));
    out[kSteps * kN * 2 + tid] = bold;
  }
}

extern "C" void kernel_launch(void* const* d_in, const int* in_sizes, int n_in,
                              void* d_out, int out_size, void* d_ws, size_t ws_size,
                              hipStream_t stream) {
  (void)in_sizes; (void)n_in; (void)d_ws; (void)ws_size; (void)out_size;
  const float* region_pars  = (const float*)d_in[0];
  const float* Wt           = (const float*)d_in[1];
  const float* g            = (const float*)d_in[2];
  const float* stimulus     = (const float*)d_in[3];
  const float* noise        = (const float*)d_in[4];
  const float* initial_cond = (const float*)d_in[5];
  const int*   lags         = (const int*)d_in[6];
  // d_in[7] (ix_lag_from) is just arange tiled == column index; unused.

  const size_t smem_bytes =
      (size_t)(kRingT * kRingPitch + kN + kN + 104) * sizeof(float);  // ~133.5 KB

  tvb_mpr_persistent<<<1, 256, smem_bytes, stream>>>(
      region_pars, Wt, g, stimulus, noise, initial_cond, lags, (float*)d_out);
}